// GCN_Net_24902220382788
// MI455X (gfx1250) — compile-verified
//
#include <hip/hip_runtime.h>
#include <math.h>

// Problem constants (match reference: N1=N2=64, N=4096, CH=64, TAU=0.05, 20 iters)
#define NNODE 4096
#define NSIDE 64
#define CHN   64
#define TAU_INV 20.0f
#define SK_ITERS 20
#define KCHUNKS 4              // split-K factor for the 4096-deep aggregation GEMM

typedef float v2f __attribute__((ext_vector_type(2)));
typedef float v8f __attribute__((ext_vector_type(8)));

// ---------------------------------------------------------------------------
// Utility: grid-stride zero fill (works for int buffers too, bit pattern 0)
// ---------------------------------------------------------------------------
__global__ void fill_zero_f32(float* __restrict__ p, int n) {
    int i = blockIdx.x * blockDim.x + threadIdx.x;
    int stride = gridDim.x * blockDim.x;
    for (; i < n; i += stride) p[i] = 0.0f;
}

// ---------------------------------------------------------------------------
// Degree accumulation: deg[d] += 1 for every non-self edge (self loop adds +1
// later as the "+1.0f" in the rsqrt kernel).
// ---------------------------------------------------------------------------
__global__ void deg_kernel(const int* __restrict__ src, const int* __restrict__ dst,
                           int E, int* __restrict__ deg) {
    int e = blockIdx.x * blockDim.x + threadIdx.x;
    int stride = gridDim.x * blockDim.x;
    for (; e < E; e += stride) {
        int s = src[e], d = dst[e];
        if (s != d) atomicAdd(&deg[d], 1);
    }
}

// dis = rsqrt(deg + 1)  (self loop contributes weight 1); also write A diag.
__global__ void dis_diag_kernel(const int* __restrict__ deg, float* __restrict__ dis,
                                float* __restrict__ A) {
    int i = blockIdx.x * blockDim.x + threadIdx.x;
    if (i >= NNODE) return;
    float d = rsqrtf((float)deg[i] + 1.0f);
    dis[i] = d;
    A[(size_t)i * NNODE + i] = d * d;   // self-loop edge, weight 1
}

// Scatter normalized off-diagonal entries: A[dst, src] = dis[src]*dis[dst].
// Edge list comes from nonzero() -> unique pairs, so plain stores suffice.
__global__ void build_A_kernel(const int* __restrict__ src, const int* __restrict__ dst,
                               int E, const float* __restrict__ dis, float* __restrict__ A) {
    int e = blockIdx.x * blockDim.x + threadIdx.x;
    int stride = gridDim.x * blockDim.x;
    for (; e < E; e += stride) {
        int s = src[e], d = dst[e];
        if (s != d) A[(size_t)d * NNODE + s] = dis[s] * dis[d];
    }
}

// ---------------------------------------------------------------------------
// Layer 0 feature prep: x0 = diag(K) (N,1).
// xw = x0 * conv_w0 (rank-1), xself = x0 * self_w0 + self_b0 (rank-1 + bias)
// ---------------------------------------------------------------------------
__global__ void layer0_prep(const float* __restrict__ Km,
                            const float* __restrict__ cw, const float* __restrict__ sw,
                            const float* __restrict__ sb,
                            float* __restrict__ xw, float* __restrict__ xself) {
    int n = blockIdx.x * blockDim.x + threadIdx.x;
    if (n >= NNODE) return;
    float xv = Km[(size_t)n * NNODE + n];
    #pragma unroll 8
    for (int o = 0; o < CHN; ++o) {
        xw[n * CHN + o]    = xv * cw[o];
        xself[n * CHN + o] = xv * sw[o] + sb[o];
    }
}

// ---------------------------------------------------------------------------
// fp32 WMMA strip GEMM:
//   P_chunk[16-row strip, 0:64] = A[strip, k0:k0+KC] @ B[k0:k0+KC, 0:64]
//
// One wave owns a 16x64 row strip = 4 independent V_WMMA_F32_16X16X4_F32
// accumulator chains; the A fragment is reused 4x per K-step.
// The K-loop is explicitly software-pipelined (prefetch depth 1, distinct
// fragment registers) so the compiler issues all 9 loads of step kk+4 before
// the 4 WMMAs of step kk -> partial s_wait_loadcnt instead of full drains.
// B has compile-time ld = 64, so B loads are immediate-offset from a single
// pointer strided by a constant per step (minimal address VALU).
//
// Split-K: blockIdx.y selects the K chunk; each chunk writes its own partial
// buffer in fixed order -> fully deterministic (no float atomics).
//
// Fragment layouts per CDNA5 ISA 7.12.2:
//   A (16x4): lanes 0-15 -> M=lane, elems K0/K1 ; lanes 16-31 -> K2/K3
//   B (4x16): elem0: lanes0-15=row K0 / lanes16-31=row K2 ; elem1: K1/K3
//   C/D     : elem v: lanes0-15 -> M=v, lanes16-31 -> M=v+8 ; N=lane&15
// Grids are exact and control flow uniform -> EXEC all-1s at every WMMA.
// ---------------------------------------------------------------------------
__global__ void wmma_strip_gemm(const float* __restrict__ A, int lda,
                                const float* __restrict__ B,
                                float* __restrict__ C, size_t chunk_stride,
                                int KC) {
    const int strip = blockIdx.x * (blockDim.x >> 5) + (threadIdx.x >> 5); // 16-row strip
    const int lane = threadIdx.x & 31;
    const int half = lane >> 4;           // 0: lanes 0-15, 1: lanes 16-31
    const int l16  = lane & 15;
    const int k0   = blockIdx.y * KC;

    v8f acc0, acc1, acc2, acc3;
    #pragma unroll
    for (int v = 0; v < 8; ++v) { acc0[v] = 0.f; acc1[v] = 0.f; acc2[v] = 0.f; acc3[v] = 0.f; }

    const float* Arow = A + (size_t)(strip * 16 + l16) * lda + k0 + 2 * half;
    // B row (k0 + 2*half), column l16; second fragment element is +CHN (next k row).
    const float* Bp = B + (size_t)(k0 + 2 * half) * CHN + l16;

    // ---- prologue: fragments for kk = 0 -----------------------------------
    v2f a, b0v, b1v, b2v, b3v;
    a.x   = Arow[0];       a.y   = Arow[1];
    b0v.x = Bp[0];         b0v.y = Bp[CHN];
    b1v.x = Bp[16];        b1v.y = Bp[CHN + 16];
    b2v.x = Bp[32];        b2v.y = Bp[CHN + 32];
    b3v.x = Bp[48];        b3v.y = Bp[CHN + 48];

    #pragma unroll 2
    for (int kk = 4; kk < KC; kk += 4) {
        // prefetch fragments for this step's successor (distinct registers)
        v2f an, b0n, b1n, b2n, b3n;
        const float* Bn = Bp + (size_t)kk * CHN;
        an.x  = Arow[kk];  an.y  = Arow[kk + 1];
        b0n.x = Bn[0];     b0n.y = Bn[CHN];
        b1n.x = Bn[16];    b1n.y = Bn[CHN + 16];
        b2n.x = Bn[32];    b2n.y = Bn[CHN + 32];
        b3n.x = Bn[48];    b3n.y = Bn[CHN + 48];
        // 4 independent WMMA chains on the already-resident fragments
        acc0 = __builtin_amdgcn_wmma_f32_16x16x4_f32(false, a, false, b0v, (short)0, acc0, false, false);
        acc1 = __builtin_amdgcn_wmma_f32_16x16x4_f32(false, a, false, b1v, (short)0, acc1, false, false);
        acc2 = __builtin_amdgcn_wmma_f32_16x16x4_f32(false, a, false, b2v, (short)0, acc2, false, false);
        acc3 = __builtin_amdgcn_wmma_f32_16x16x4_f32(false, a, false, b3v, (short)0, acc3, false, false);
        a = an; b0v = b0n; b1v = b1n; b2v = b2n; b3v = b3n;
    }
    // ---- epilogue: last K-step --------------------------------------------
    acc0 = __builtin_amdgcn_wmma_f32_16x16x4_f32(false, a, false, b0v, (short)0, acc0, false, false);
    acc1 = __builtin_amdgcn_wmma_f32_16x16x4_f32(false, a, false, b1v, (short)0, acc1, false, false);
    acc2 = __builtin_amdgcn_wmma_f32_16x16x4_f32(false, a, false, b2v, (short)0, acc2, false, false);
    acc3 = __builtin_amdgcn_wmma_f32_16x16x4_f32(false, a, false, b3v, (short)0, acc3, false, false);

    float* Cout = C + blockIdx.y * chunk_stride;
    #pragma unroll
    for (int v = 0; v < 8; ++v) {
        size_t r = (size_t)(strip * 16 + v + 8 * half) * CHN + l16;
        Cout[r]      = acc0[v];
        Cout[r + 16] = acc1[v];
        Cout[r + 32] = acc2[v];
        Cout[r + 48] = acc3[v];
    }
}

// ---------------------------------------------------------------------------
// Deterministic split-K reduce fused with the GCN layer combine:
//   xo[n,c] = xself[n,c] + sum_chunks P[c][n,c] + cb[c] (+ sb[c] for layers>=1)
// ---------------------------------------------------------------------------
__global__ void combine_layer(const float* __restrict__ xself,
                              const float* __restrict__ P, size_t chunk_stride,
                              const float* __restrict__ sb_or_null,
                              const float* __restrict__ cb,
                              float* __restrict__ xo) {
    int i = blockIdx.x * blockDim.x + threadIdx.x;
    if (i >= NNODE * CHN) return;
    int c = i & (CHN - 1);
    float s = xself[i];
    #pragma unroll
    for (int q = 0; q < KCHUNKS; ++q) s += P[q * chunk_stride + i];
    s += cb[c];
    if (sb_or_null) s += sb_or_null[c];
    xo[i] = s;
}

// ---------------------------------------------------------------------------
// Row dot: out[n] = dot(X[n,:64], w) + b[0]   (classifier / final head)
// ---------------------------------------------------------------------------
__global__ void rowdot64(const float* __restrict__ X, const float* __restrict__ w,
                         const float* __restrict__ b, float* __restrict__ out) {
    int n = blockIdx.x * blockDim.x + threadIdx.x;
    if (n >= NNODE) return;
    float s = 0.0f;
    #pragma unroll 8
    for (int c = 0; c < CHN; ++c) s += X[n * CHN + c] * w[c];
    out[n] = s + b[0];
}

// ---------------------------------------------------------------------------
// Log-domain Sinkhorn on a 64x64 matrix. One block of 64 threads; matrix in
// LDS (padded stride 65 to avoid bank conflicts on the row pass). Even iters
// normalize rows (axis=1), odd iters columns (axis=0). Output = exp(log_s).
// transpose_in: read in[j*64+i] (for final scores.reshape(n2,n1).T).
// ---------------------------------------------------------------------------
__global__ void sinkhorn64(const float* __restrict__ in, float* __restrict__ out,
                           int transpose_in) {
    __shared__ float ls[NSIDE][NSIDE + 1];
    int t = threadIdx.x;
    for (int j = 0; j < NSIDE; ++j) {
        float v = transpose_in ? in[j * NSIDE + t] : in[t * NSIDE + j];
        ls[t][j] = v * TAU_INV;
    }
    __syncthreads();
    for (int it = 0; it < SK_ITERS; ++it) {
        if ((it & 1) == 0) {                 // row logsumexp: thread t owns row t
            float m = -INFINITY;
            for (int j = 0; j < NSIDE; ++j) m = fmaxf(m, ls[t][j]);
            float s = 0.0f;
            for (int j = 0; j < NSIDE; ++j) s += expf(ls[t][j] - m);
            float lse = m + logf(s);
            for (int j = 0; j < NSIDE; ++j) ls[t][j] -= lse;
        } else {                             // column logsumexp: thread t owns col t
            float m = -INFINITY;
            for (int i = 0; i < NSIDE; ++i) m = fmaxf(m, ls[i][t]);
            float s = 0.0f;
            for (int i = 0; i < NSIDE; ++i) s += expf(ls[i][t] - m);
            float lse = m + logf(s);
            for (int i = 0; i < NSIDE; ++i) ls[i][t] -= lse;
        }
        __syncthreads();
    }
    for (int j = 0; j < NSIDE; ++j) out[t * NSIDE + j] = expf(ls[t][j]);
}

// x[n,c] = xo[n,c] + sk_flat[n]   (broadcast Sinkhorn result over channels)
__global__ void add_sk(const float* __restrict__ xo, const float* __restrict__ sk,
                       float* __restrict__ x) {
    int i = blockIdx.x * blockDim.x + threadIdx.x;
    if (i < NNODE * CHN) x[i] = xo[i] + sk[i >> 6];
}

// ---------------------------------------------------------------------------
// Host-side orchestration
// ---------------------------------------------------------------------------
extern "C" void kernel_launch(void* const* d_in, const int* in_sizes, int n_in,
                              void* d_out, int out_size, void* d_ws, size_t ws_size,
                              hipStream_t stream) {
    const float* Km = (const float*)d_in[0];
    const int*   ei = (const int*)d_in[1];
    const int E = in_sizes[1] / 2;
    const int* src = ei;          // edge_index[0]
    const int* dst = ei + E;      // edge_index[1]

    const float* cw[3] = {(const float*)d_in[4],  (const float*)d_in[10], (const float*)d_in[16]};
    const float* cb[3] = {(const float*)d_in[5],  (const float*)d_in[11], (const float*)d_in[17]};
    const float* sw[3] = {(const float*)d_in[6],  (const float*)d_in[12], (const float*)d_in[18]};
    const float* sb[3] = {(const float*)d_in[7],  (const float*)d_in[13], (const float*)d_in[19]};
    const float* kw[3] = {(const float*)d_in[8],  (const float*)d_in[14], (const float*)d_in[20]};
    const float* kb[3] = {(const float*)d_in[9],  (const float*)d_in[15], (const float*)d_in[21]};
    const float* fw = (const float*)d_in[22];
    const float* fb = (const float*)d_in[23];

    // Workspace layout (fp32 words): A (67MB) is L2-resident after first touch.
    const size_t XSZ = (size_t)NNODE * CHN;            // 262144
    float* A     = (float*)d_ws;                       // 4096*4096
    float* x     = A + (size_t)NNODE * NNODE;          // N x 64 current features
    float* xw    = x + XSZ;                            // N x 64  (x @ conv_w)
    float* xself = xw + XSZ;                           // N x 64  (x @ self_w)
    float* xo    = xself + XSZ;                        // N x 64  layer output
    float* P     = xo + XSZ;                           // KCHUNKS partials, N x 64 each
    float* svec  = P + (size_t)KCHUNKS * XSZ;          // 4096
    float* smat  = svec + NNODE;                       // 4096
    float* dis   = smat + NNODE;                       // 4096
    int*   deg   = (int*)(dis + NNODE);                // 4096

    // ---- Build dense normalized adjacency ---------------------------------
    fill_zero_f32<<<4096, 256, 0, stream>>>(A, NNODE * NNODE);
    fill_zero_f32<<<16,   256, 0, stream>>>((float*)deg, NNODE);
    deg_kernel<<<1024, 256, 0, stream>>>(src, dst, E, deg);
    dis_diag_kernel<<<(NNODE + 255) / 256, 256, 0, stream>>>(deg, dis, A);
    build_A_kernel<<<1024, 256, 0, stream>>>(src, dst, E, dis, A);

    // Strip-GEMM grids: 8 waves/block, each wave owns a 16x64 row strip.
    const dim3 gridBig(NNODE / 128, KCHUNKS);          // split-K aggregation GEMM
    const dim3 gridSmall(NNODE / 128, 1);              // feature GEMMs (K=64)

    for (int l = 0; l < 3; ++l) {
        if (l == 0) {
            layer0_prep<<<(NNODE + 255) / 256, 256, 0, stream>>>(Km, cw[0], sw[0], sb[0], xw, xself);
        } else {
            // xw = x @ cw         (4096x64 @ 64x64)
            wmma_strip_gemm<<<gridSmall, 256, 0, stream>>>(x, CHN, cw[l], xw, 0, CHN);
            // xself = x @ sw      (bias sb folded into combine)
            wmma_strip_gemm<<<gridSmall, 256, 0, stream>>>(x, CHN, sw[l], xself, 0, CHN);
        }
        // P[q] = A[:, q-chunk] @ xw[q-chunk, :]   (4096x4096 @ 4096x64, split-K)
        wmma_strip_gemm<<<gridBig, 256, 0, stream>>>(A, NNODE, xw, P, XSZ,
                                                     NNODE / KCHUNKS);
        // xo = xself + sum_q P[q] + cb (+ sb for l>=1)  -- deterministic reduce
        combine_layer<<<(NNODE * CHN + 255) / 256, 256, 0, stream>>>(
            xself, P, XSZ, (l == 0) ? nullptr : sb[l], cb[l], xo);
        // classifier -> Sinkhorn -> broadcast add
        rowdot64<<<(NNODE + 255) / 256, 256, 0, stream>>>(xo, kw[l], kb[l], svec);
        sinkhorn64<<<1, NSIDE, 0, stream>>>(svec, smat, 0);
        add_sk<<<(NNODE * CHN + 255) / 256, 256, 0, stream>>>(xo, smat, x);
    }

    // ---- Final head: scores -> reshape(n2,n1).T -> Sinkhorn ---------------
    rowdot64<<<(NNODE + 255) / 256, 256, 0, stream>>>(x, fw, fb, svec);
    sinkhorn64<<<1, NSIDE, 0, stream>>>(svec, (float*)d_out, 1);
}